// Attend_26121991094586
// MI455X (gfx1250) — compile-verified
//
#include <hip/hip_runtime.h>
#include <hip/hip_bf16.h>

typedef __attribute__((ext_vector_type(2))) float v2f;
typedef __attribute__((ext_vector_type(8))) float v8f;
typedef int vi4 __attribute__((vector_size(16)));   // matches builtin param type

#define B_ 32
#define N_ 8192
#define D_ 128
#define CHUNKS 64            // N_ / ROWS_PER_BLOCK
#define ROWS_PER_BLOCK 128
#define WAVES 8
#define ROWS_PER_WAVE 16
#define VS_STRIDE 132        // 128 + 4 pad: rows 16B-aligned, conflict-free gathers
#define PART_STRIDE 130      // m, l, ctx[128]

#if __has_builtin(__builtin_amdgcn_global_load_async_to_lds_b128) && \
    __has_builtin(__builtin_amdgcn_s_wait_asynccnt)
#define USE_ASYNC_LDS 1
#else
#define USE_ASYNC_LDS 0
#endif

// Single shared struct: q_s at offset 0 so the hot-loop q loads fold into
// small immediate DS offsets (ds_load_2addr_b64) instead of per-load v_or.
struct SharedMem {
    float q_s[D_];                                        // offset 0
    float redm[WAVES];
    float redl[WAVES];
    __align__(16) float redc[WAVES][D_];                  // offset 576 (16B aligned)
    __align__(16) float vs[WAVES][ROWS_PER_WAVE][VS_STRIDE];
};

// ---------------------------------------------------------------------------
// Kernel 1: per-(batch, chunk) flash-decode partial.
// Each wave owns 16 rows of V. Scores via V_WMMA_F32_16X16X4_F32 with q
// broadcast into all 16 A-rows; V tile as B. acc[0] then holds
// score[row lane%16] in every lane (M rows identical by construction).
// ---------------------------------------------------------------------------
__global__ __launch_bounds__(256)
void attend_partial(const float* __restrict__ q,
                    const float* __restrict__ v,
                    float* __restrict__ attn_raw,   // d_out[0 .. B*N): raw scores
                    float* __restrict__ part)       // [B][CHUNKS][130]
{
    __shared__ SharedMem smem;

    const int tid  = threadIdx.x;
    const int wv   = tid >> 5;
    const int lane = tid & 31;
    const int b    = blockIdx.x / CHUNKS;
    const int c    = blockIdx.x % CHUNKS;
    const int row0 = c * ROWS_PER_BLOCK + wv * ROWS_PER_WAVE;  // first row of this wave

    if (tid < D_) smem.q_s[tid] = q[b * D_ + tid];

    // Stage this wave's 16 rows into LDS, one full 512B row per instruction.
    const float* vg = v + ((size_t)b * N_ + row0) * D_;
    __builtin_prefetch(vg, 0, 0);   // global_prefetch_b8
#if USE_ASYNC_LDS
#pragma unroll
    for (int i = 0; i < ROWS_PER_WAVE; ++i) {
        const float* src = vg + (size_t)i * D_ + 4 * lane;
        __builtin_amdgcn_global_load_async_to_lds_b128(
            (__attribute__((address_space(1))) vi4*)src,
            (__attribute__((address_space(3))) vi4*)&smem.vs[wv][i][4 * lane],
            /*offset=*/0, /*cpol=*/0);
    }
    __builtin_amdgcn_s_wait_asynccnt(0);
#else
#pragma unroll
    for (int i = 0; i < ROWS_PER_WAVE; ++i) {
        const float4 t = *((const float4*)(vg + (size_t)i * D_) + lane);
        *(float4*)&smem.vs[wv][i][4 * lane] = t;
    }
#endif
    __syncthreads();

    // WMMA scores: A = q broadcast to 16 rows, B = V tile (16 rows x K).
    // Per-lane float2 operands at K offset k0 + 2*(lane>>4).
    const int    koff = (lane >> 4) << 1;     // 0 or 2
    const float* vrow = &smem.vs[wv][lane & 15][0];

    v8f acc = {0.f, 0.f, 0.f, 0.f, 0.f, 0.f, 0.f, 0.f};
#pragma unroll
    for (int k = 0; k < D_; k += 4) {
        v2f a  = *(const v2f*)&smem.q_s[k + koff];
        v2f bb = *(const v2f*)&vrow[k + koff];
        acc = __builtin_amdgcn_wmma_f32_16x16x4_f32(
            /*neg_a=*/false, a, /*neg_b=*/false, bb,
            /*c_mod=*/(short)0, acc, /*reuse_a=*/false, /*reuse_b=*/false);
    }
    const float s = acc[0];   // score of row (lane & 15), duplicated in upper half-wave

    // 16-lane-group reductions (upper half holds duplicates, masks < 16 are safe).
    float m = s;
#pragma unroll
    for (int off = 8; off >= 1; off >>= 1)
        m = fmaxf(m, __shfl_xor(m, off, 32));
    const float w = __expf(s - m);
    float l = w;
#pragma unroll
    for (int off = 8; off >= 1; off >>= 1)
        l += __shfl_xor(l, off, 32);

    if (lane < 16)
        attn_raw[(size_t)b * N_ + row0 + lane] = s;

    // Context partial: each lane owns 4 D-columns (32 lanes * 4 = 128).
    float4 ctx = make_float4(0.f, 0.f, 0.f, 0.f);
#pragma unroll
    for (int j = 0; j < ROWS_PER_WAVE; ++j) {
        const float  wj = __shfl(w, j, 32);
        const float4 vv = *(const float4*)&smem.vs[wv][j][4 * lane];
        ctx.x += wj * vv.x;
        ctx.y += wj * vv.y;
        ctx.z += wj * vv.z;
        ctx.w += wj * vv.w;
    }

    // Cross-wave combine within the block.
    *(float4*)&smem.redc[wv][4 * lane] = ctx;
    if (lane == 0) { smem.redm[wv] = m; smem.redl[wv] = l; }
    __syncthreads();

    float* p = part + ((size_t)b * CHUNKS + c) * PART_STRIDE;
    if (tid < D_) {
        float mb = smem.redm[0];
#pragma unroll
        for (int i = 1; i < WAVES; ++i) mb = fmaxf(mb, smem.redm[i]);
        float ctxb = 0.f, lb = 0.f;
#pragma unroll
        for (int i = 0; i < WAVES; ++i) {
            const float sc = __expf(smem.redm[i] - mb);
            ctxb += smem.redc[i][tid] * sc;
            lb   += smem.redl[i] * sc;
        }
        p[2 + tid] = ctxb;
        if (tid == 0) { p[0] = mb; p[1] = lb; }
    }
}

// ---------------------------------------------------------------------------
// Kernel 2: merge 64 chunk-partials per batch; write context and (m, 1/l).
// ---------------------------------------------------------------------------
__global__ __launch_bounds__(128)
void attend_combine(const float* __restrict__ part,
                    float* __restrict__ ctx_out,   // d_out + B*N
                    float* __restrict__ stats)     // [B][2] = {m, 1/l}
{
    const int b = blockIdx.x;
    const int d = threadIdx.x;   // 0..127
    const float* p = part + (size_t)b * CHUNKS * PART_STRIDE;

    float mb = p[0];
    for (int c = 1; c < CHUNKS; ++c)
        mb = fmaxf(mb, p[(size_t)c * PART_STRIDE]);

    float lb = 0.f, ctx = 0.f;
    for (int c = 0; c < CHUNKS; ++c) {
        const float* pc = p + (size_t)c * PART_STRIDE;
        const float  sc = __expf(pc[0] - mb);
        lb  += pc[1] * sc;
        ctx += pc[2 + d] * sc;
    }

    ctx_out[b * D_ + d] = ctx / lb;
    if (d == 0) { stats[2 * b] = mb; stats[2 * b + 1] = 1.0f / lb; }
}

// ---------------------------------------------------------------------------
// Kernel 3: normalize raw scores in place -> attn.
// ---------------------------------------------------------------------------
__global__ __launch_bounds__(256)
void attend_norm(float* __restrict__ attn, const float* __restrict__ stats)
{
    const int i = blockIdx.x * 256 + threadIdx.x;   // grid sized exactly B*N
    const int b = i >> 13;                          // N_ = 8192
    const float s = attn[i];
    attn[i] = __expf(s - stats[2 * b]) * stats[2 * b + 1];
}

// ---------------------------------------------------------------------------
extern "C" void kernel_launch(void* const* d_in, const int* in_sizes, int n_in,
                              void* d_out, int out_size, void* d_ws, size_t ws_size,
                              hipStream_t stream) {
    (void)in_sizes; (void)n_in; (void)out_size; (void)ws_size;

    const float* q = (const float*)d_in[0];   // [B,1,D]
    const float* v = (const float*)d_in[1];   // [B,N,D]

    float* out  = (float*)d_out;
    float* attn = out;                        // B*N floats
    float* ctx  = out + (size_t)B_ * N_;      // B*D floats

    float* part  = (float*)d_ws;                                   // B*CHUNKS*130
    float* stats = part + (size_t)B_ * CHUNKS * PART_STRIDE;       // B*2

    attend_partial<<<dim3(B_ * CHUNKS), dim3(256), 0, stream>>>(q, v, attn, part);
    attend_combine<<<dim3(B_), dim3(128), 0, stream>>>(part, ctx, stats);
    attend_norm<<<dim3((B_ * N_) / 256), dim3(256), 0, stream>>>(attn, stats);
}